// QNStepTD_30073361007336
// MI455X (gfx1250) — compile-verified
//
#include <hip/hip_runtime.h>
#include <hip/hip_bf16.h>

// CDNA5 / gfx1250 n-step TD loss.
// Memory-bound: ~50 MB traffic -> ~2-7 us at 23.3 TB/s. One lane per sample,
// coalesced reward/weight/action streams, dword gathers for Q(s,a).
// The einsum('t,tb->b') discount dot is computed with V_WMMA_F32_16X16X4_F32:
//   A (16x4)  = broadcast discount row [1, g, g^2, g^3]
//   B (4x16)  = reward rows r0..r3 for 16 samples
//   C (16x16) = tail g^4*r4 + g^5*Q'*(1-done) broadcast down columns
// Two tiles per wave cover all 32 lanes. Deterministic two-pass reduction
// (block partials in d_ws, fixed-order tree reduce) -> loss.

typedef __attribute__((ext_vector_type(2))) float v2f;
typedef __attribute__((ext_vector_type(8))) float v8f;

#define BDIM 256

__global__ __launch_bounds__(BDIM) void td_main_kernel(
    const float* __restrict__ q, const float* __restrict__ nq,
    const int* __restrict__ act, const int* __restrict__ nact,
    const float* __restrict__ rew, const unsigned char* __restrict__ done,
    const float* __restrict__ w, const float* __restrict__ gammap,
    float* __restrict__ out_td, float* __restrict__ partials, int Bn)
{
    const int b    = blockIdx.x * BDIM + threadIdx.x;   // grid exactly covers B
    const int lane = threadIdx.x & 31;

    const float g  = gammap[0];
    const float g2 = g * g;
    const float g3 = g2 * g;
    const float g4 = g2 * g2;
    const float g5 = g4 * g;

    // Coalesced streams: reward is (T,B) so each t-slab is contiguous in b.
    const float r0 = rew[0 * (size_t)Bn + b];
    const float r1 = rew[1 * (size_t)Bn + b];
    const float r2 = rew[2 * (size_t)Bn + b];
    const float r3 = rew[3 * (size_t)Bn + b];
    const float r4 = rew[4 * (size_t)Bn + b];

    const int   a   = act[b];
    const int   na  = nact[b];
    const float qsa = q [(size_t)b * 16 + a];    // row = 64B; gather pulls the
    const float nqa = nq[(size_t)b * 16 + na];   // same sectors as a row read
    const float notdone = 1.0f - (float)done[b]; // jnp.bool_ -> 1 byte
    const float tail = g4 * r4 + g5 * nqa * notdone;

    // Cross-half operand marshalling (lane <-> lane^16), wave32.
    const float xr0   = __shfl_xor(r0,   16, 32);
    const float xr1   = __shfl_xor(r1,   16, 32);
    const float xr2   = __shfl_xor(r2,   16, 32);
    const float xr3   = __shfl_xor(r3,   16, 32);
    const float xtail = __shfl_xor(tail, 16, 32);

    const bool lo = lane < 16;

    // A 16x4 f32 layout: lanes 0-15 hold K=0 (v0), K=1 (v1); lanes 16-31 hold K=2,3.
    v2f A;
    A.x = lo ? 1.0f : g2;
    A.y = lo ? g    : g3;

    // B 4x16 f32 layout: lanes 0-15 hold K=0 (v0), K=1 (v1); lanes 16-31 hold K=2,3.
    // Tile0: samples of lanes 0-15.  Tile1: samples of lanes 16-31.
    v2f B0; B0.x = lo ? r0  : xr2; B0.y = lo ? r1  : xr3;
    v2f B1; B1.x = lo ? xr0 : r2;  B1.y = lo ? xr1 : r3;

    // C/D 16x16: VGPR j, lanes 0-15 -> (M=j, N=lane); lanes 16-31 -> (M=j+8, N=lane-16).
    // All rows identical, so broadcast tail[N] into every C VGPR.
    const float c0v = lo ? tail  : xtail;
    const float c1v = lo ? xtail : tail;
    v8f C0 = {c0v, c0v, c0v, c0v, c0v, c0v, c0v, c0v};
    v8f C1 = {c1v, c1v, c1v, c1v, c1v, c1v, c1v, c1v};

    // D = A x B + C  (EXEC is all ones here: no divergence above)
    v8f D0 = __builtin_amdgcn_wmma_f32_16x16x4_f32(
        false, A, false, B0, (short)0, C0, false, false);
    v8f D1 = __builtin_amdgcn_wmma_f32_16x16x4_f32(
        false, A, false, B1, (short)0, C1, false, false);

    // Every lane reads its own sample's target from D VGPR 0.
    const float target = lo ? D0[0] : D1[0];
    const float diff   = qsa - target;
    const float td     = diff * diff;
    out_td[b] = td;

    // Deterministic block reduction of td * weight.
    float p = td * w[b];
    #pragma unroll
    for (int off = 16; off > 0; off >>= 1)
        p += __shfl_down(p, off, 32);

    __shared__ float sm[BDIM / 32];
    if (lane == 0) sm[threadIdx.x >> 5] = p;
    __syncthreads();
    if (threadIdx.x == 0) {
        float s = 0.0f;
        #pragma unroll
        for (int i = 0; i < BDIM / 32; ++i) s += sm[i];
        partials[blockIdx.x] = s;
    }
}

__global__ __launch_bounds__(256) void td_finalize_kernel(
    const float* __restrict__ partials, float* __restrict__ out_loss,
    int nblk, float invB)
{
    __shared__ float sm[256];
    float s = 0.0f;
    for (int i = threadIdx.x; i < nblk; i += 256)   // fixed order -> deterministic
        s += partials[i];
    sm[threadIdx.x] = s;
    __syncthreads();
    #pragma unroll
    for (int off = 128; off > 0; off >>= 1) {
        if (threadIdx.x < off) sm[threadIdx.x] += sm[threadIdx.x + off];
        __syncthreads();
    }
    if (threadIdx.x == 0) out_loss[0] = sm[0] * invB;
}

extern "C" void kernel_launch(void* const* d_in, const int* in_sizes, int n_in,
                              void* d_out, int out_size, void* d_ws, size_t ws_size,
                              hipStream_t stream) {
    const float*         q    = (const float*)        d_in[0]; // (B,16) f32
    const float*         nq   = (const float*)        d_in[1]; // (B,16) f32
    const int*           act  = (const int*)          d_in[2]; // (B,)   i32
    const int*           nact = (const int*)          d_in[3]; // (B,)   i32
    const float*         rew  = (const float*)        d_in[4]; // (T,B)  f32
    const unsigned char* done = (const unsigned char*)d_in[5]; // (B,)   bool (1B)
    const float*         w    = (const float*)        d_in[6]; // (B,)   f32
    const float*         gam  = (const float*)        d_in[7]; // ()     f32

    const int Bn   = in_sizes[2];      // B = 1048576
    const int nblk = Bn / BDIM;        // exact: B is a multiple of 256

    float* out      = (float*)d_out;   // [0] = loss, [1..B] = td_error_per_sample
    float* partials = (float*)d_ws;    // nblk floats of scratch

    td_main_kernel<<<nblk, BDIM, 0, stream>>>(
        q, nq, act, nact, rew, done, w, gam, out + 1, partials, Bn);

    td_finalize_kernel<<<1, 256, 0, stream>>>(
        partials, out, nblk, 1.0f / (float)Bn);
}